// HGNN_3951369912999
// MI455X (gfx1250) — compile-verified
//
#include <hip/hip_runtime.h>
#include <hip/hip_bf16.h>
#include <stddef.h>

// ---------------------------------------------------------------------------
// HGNN forward + diagnostics for MI455X (gfx1250), f32 WMMA (16x16x4) path.
// Branch-free, software-pipelined GEMM; compile-time strides; explicit
// global-address-space fragment pointers (global_load, not flat_load).
// ---------------------------------------------------------------------------

typedef __attribute__((ext_vector_type(2))) float v2f;
typedef __attribute__((ext_vector_type(8))) float v8f;

typedef float __attribute__((address_space(1))) gfloat;

__device__ __forceinline__ const gfloat* as_global(const float* p) {
  return (const gfloat*)(unsigned long long)p;
}
__device__ __forceinline__ gfloat* as_global_mut(float* p) {
  return (gfloat*)(unsigned long long)p;
}

#define N_NODES 4096
#define N_EDGES 2048
#define IN_CH   512
#define HIDW    512
#define NCLS    40
#define EPSF    1e-10f

// ------------------------- block reduction helpers -------------------------
template<int NT>
__device__ inline float blk_sum(float v, float* red) {
  const int tid = threadIdx.x;
  red[tid] = v; __syncthreads();
  #pragma unroll
  for (int s = NT / 2; s > 0; s >>= 1) {
    if (tid < s) red[tid] += red[tid + s];
    __syncthreads();
  }
  float r = red[0]; __syncthreads();
  return r;
}

// ------------------------------ WMMA GEMM ----------------------------------
// MODE 0 (NN): C[M,F] = A(row-major MxK, LDA) * B(row-major KxF, LDB)
// MODE 1 (TN): C[M,F] = A_st^T * (diag(bscale) B); A_st stored KxM (LDA = M dim)
// MODE 2 (NT): C[M,F] = A * B_st^T; B_st stored FxK (LDB = K dim); fused ReLU
//
// Compile-time strides -> constant pointer increments + constant instruction
// offsets. Out-of-range columns (NCOLS=40 case) use clamped load addresses
// and are never stored. Inner loop is branch-free and EXEC-uniform.
template<int MODE, bool RELU, int LDA, int LDB, int LDC, int NCOLS>
__global__ __launch_bounds__(256)
void hgnn_wmma_gemm(const float* __restrict__ A, const float* __restrict__ B,
                    float* __restrict__ C, const float* __restrict__ bscale,
                    int K)
{
  const int lane = threadIdx.x & 31;
  const int wave = threadIdx.x >> 5;
  const int hv   = lane >> 4;     // 0: K=0,1 half ; 1: K=2,3 half
  const int lid  = lane & 15;
  const int m0   = (blockIdx.x * 8 + wave) * 16;
  const int nb   = blockIdx.y * 64;
  const int arow = m0 + lid;      // A row (modes 0/2) or A_st column (mode 1)

  // clamped column indices for loads; real n used for stores
  int ncl[4];
  #pragma unroll
  for (int j = 0; j < 4; ++j) {
    const int n = nb + 16 * j + lid;
    ncl[j] = (n < NCOLS) ? n : (NCOLS - 1);
  }

  // ---- fragment pointers (global AS), constant step per K-iteration -------
  const gfloat* pa = as_global((MODE == 1) ? A + (size_t)(2 * hv) * LDA + arow
                                           : A + (size_t)arow * LDA + 2 * hv);
  const size_t pa_step = (MODE == 1) ? (size_t)4 * LDA : (size_t)4;
  const int    pa_off  = (MODE == 1) ? LDA : 1;   // a.y at pa[pa_off]

  const gfloat* pb0;
  const gfloat* pb1;
  const gfloat* pb2;
  const gfloat* pb3;
  if (MODE == 2) {
    pb0 = as_global(B + (size_t)ncl[0] * LDB + 2 * hv);
    pb1 = as_global(B + (size_t)ncl[1] * LDB + 2 * hv);
    pb2 = as_global(B + (size_t)ncl[2] * LDB + 2 * hv);
    pb3 = as_global(B + (size_t)ncl[3] * LDB + 2 * hv);
  } else {
    pb0 = as_global(B + (size_t)(2 * hv) * LDB + ncl[0]);
    pb1 = as_global(B + (size_t)(2 * hv) * LDB + ncl[1]);
    pb2 = as_global(B + (size_t)(2 * hv) * LDB + ncl[2]);
    pb3 = as_global(B + (size_t)(2 * hv) * LDB + ncl[3]);
  }
  const size_t pb_step = (MODE == 2) ? (size_t)4 : (size_t)4 * LDB;
  const int    pb_off  = (MODE == 2) ? 1 : LDB;   // b.y at pb[pb_off]

  const gfloat* ps = as_global(bscale) + 2 * hv;  // MODE 1 only

  v8f c0 = {}, c1 = {}, c2 = {}, c3 = {};

  v2f a, b0, b1, b2, b3;
  // prologue load
  if (MODE == 2) {
    a  = *(const __attribute__((address_space(1))) v2f*)pa;
    b0 = *(const __attribute__((address_space(1))) v2f*)pb0;
    b1 = *(const __attribute__((address_space(1))) v2f*)pb1;
    b2 = *(const __attribute__((address_space(1))) v2f*)pb2;
    b3 = *(const __attribute__((address_space(1))) v2f*)pb3;
  } else {
    a.x = pa[0]; a.y = pa[pa_off];
    b0.x = pb0[0]; b0.y = pb0[pb_off];
    b1.x = pb1[0]; b1.y = pb1[pb_off];
    b2.x = pb2[0]; b2.y = pb2[pb_off];
    b3.x = pb3[0]; b3.y = pb3[pb_off];
  }
  if (MODE == 1) {
    const float s0 = ps[0], s1 = ps[1];
    b0.x *= s0; b0.y *= s1; b1.x *= s0; b1.y *= s1;
    b2.x *= s0; b2.y *= s1; b3.x *= s0; b3.y *= s1;
    ps += 4;
  }
  pa += pa_step;
  pb0 += pb_step; pb1 += pb_step; pb2 += pb_step; pb3 += pb_step;

  #pragma unroll 2
  for (int k = 0; k < K - 4; k += 4) {
    v2f an, bn0, bn1, bn2, bn3;
    if (MODE == 2) {
      an  = *(const __attribute__((address_space(1))) v2f*)pa;
      bn0 = *(const __attribute__((address_space(1))) v2f*)pb0;
      bn1 = *(const __attribute__((address_space(1))) v2f*)pb1;
      bn2 = *(const __attribute__((address_space(1))) v2f*)pb2;
      bn3 = *(const __attribute__((address_space(1))) v2f*)pb3;
    } else {
      an.x = pa[0]; an.y = pa[pa_off];
      bn0.x = pb0[0]; bn0.y = pb0[pb_off];
      bn1.x = pb1[0]; bn1.y = pb1[pb_off];
      bn2.x = pb2[0]; bn2.y = pb2[pb_off];
      bn3.x = pb3[0]; bn3.y = pb3[pb_off];
    }
    if (MODE == 1) {
      const float s0 = ps[0], s1 = ps[1];
      bn0.x *= s0; bn0.y *= s1; bn1.x *= s0; bn1.y *= s1;
      bn2.x *= s0; bn2.y *= s1; bn3.x *= s0; bn3.y *= s1;
      ps += 4;
    }
    pa += pa_step;
    pb0 += pb_step; pb1 += pb_step; pb2 += pb_step; pb3 += pb_step;

    c0 = __builtin_amdgcn_wmma_f32_16x16x4_f32(false, a, false, b0, (short)0, c0, false, false);
    c1 = __builtin_amdgcn_wmma_f32_16x16x4_f32(false, a, false, b1, (short)0, c1, false, false);
    c2 = __builtin_amdgcn_wmma_f32_16x16x4_f32(false, a, false, b2, (short)0, c2, false, false);
    c3 = __builtin_amdgcn_wmma_f32_16x16x4_f32(false, a, false, b3, (short)0, c3, false, false);
    a = an; b0 = bn0; b1 = bn1; b2 = bn2; b3 = bn3;
  }
  c0 = __builtin_amdgcn_wmma_f32_16x16x4_f32(false, a, false, b0, (short)0, c0, false, false);
  c1 = __builtin_amdgcn_wmma_f32_16x16x4_f32(false, a, false, b1, (short)0, c1, false, false);
  c2 = __builtin_amdgcn_wmma_f32_16x16x4_f32(false, a, false, b2, (short)0, c2, false, false);
  c3 = __builtin_amdgcn_wmma_f32_16x16x4_f32(false, a, false, b3, (short)0, c3, false, false);

  gfloat* Cg = as_global_mut(C);
  const v8f accs[4] = { c0, c1, c2, c3 };
  #pragma unroll
  for (int j = 0; j < 4; ++j) {
    const int n = nb + 16 * j + lid;
    if (n < NCOLS) {
      #pragma unroll
      for (int r = 0; r < 8; ++r) {
        const int m = m0 + r + 8 * hv;
        float vv = accs[j][r];
        if (RELU) vv = fmaxf(vv, 0.0f);
        Cg[(size_t)m * LDC + n] = vv;
      }
    }
  }
}

// ------------------------- degrees / scalings ------------------------------
__global__ __launch_bounds__(256)
void hgnn_rowdeg(const float* __restrict__ H, float* __restrict__ dv,
                 float* __restrict__ rs) {
  __shared__ float red[256];
  const int row = blockIdx.x, tid = threadIdx.x;
  float acc = 0.0f;
  for (int e = tid; e < N_EDGES; e += 256) acc += H[(size_t)row * N_EDGES + e];
  acc = blk_sum<256>(acc, red);
  if (tid == 0) { dv[row] = acc; rs[row] = 1.0f / sqrtf(acc + EPSF); }
}

__global__ __launch_bounds__(256)
void hgnn_coldeg(const float* __restrict__ H, float* __restrict__ de) {
  const int e = blockIdx.x * 256 + threadIdx.x;
  if (e < N_EDGES) {
    float acc = 0.0f;
    for (int n = 0; n < N_NODES; ++n) acc += H[(size_t)n * N_EDGES + e];
    de[e] = acc;
  }
}

// s[n] = (sum_f x[n,f]^2) / (dv[n]+eps)  == sum_f y^2
__global__ __launch_bounds__(128)
void hgnn_rowsq(const float* __restrict__ X, const float* __restrict__ rs,
                float* __restrict__ s, int F) {
  __shared__ float red[128];
  const int row = blockIdx.x, tid = threadIdx.x;
  float acc = 0.0f;
  for (int f = tid; f < F; f += 128) {
    float x = X[(size_t)row * F + f];
    acc += x * x;
  }
  acc = blk_sum<128>(acc, red);
  if (tid == 0) { float r = rs[row]; s[row] = acc * r * r; }
}

// a[e] = sum_n H[n,e] * s[n]
__global__ __launch_bounds__(256)
void hgnn_edge_a(const float* __restrict__ H, const float* __restrict__ s,
                 float* __restrict__ a) {
  const int e = blockIdx.x * 256 + threadIdx.x;
  if (e < N_EDGES) {
    float acc = 0.0f;
    for (int n = 0; n < N_NODES; ++n) acc += H[(size_t)n * N_EDGES + e] * s[n];
    a[e] = acc;
  }
}

// per-edge contribution: (b*a - ||M_e||^2)/max(b,eps) if b>0 else 0
__global__ __launch_bounds__(256)
void hgnn_edge_energy(const float* __restrict__ M, const float* __restrict__ a,
                      const float* __restrict__ de, int ldm, int ncols,
                      float* __restrict__ pb) {
  __shared__ float red[256];
  const int e = blockIdx.x, tid = threadIdx.x;
  float acc = 0.0f;
  for (int f = tid; f < ncols; f += 256) {
    float m = M[(size_t)e * ldm + f];
    acc += m * m;
  }
  acc = blk_sum<256>(acc, red);
  if (tid == 0) {
    float b = de[e];
    pb[e] = (b > 0.0f) ? (b * a[e] - acc) / fmaxf(b, EPSF) : 0.0f;
  }
}

// --------------------- deterministic two-stage reductions ------------------
__global__ __launch_bounds__(256)
void hgnn_dot_partial(const float* __restrict__ x, const float* __restrict__ y,
                      long n, float* __restrict__ pb) {
  __shared__ float red[256];
  const long i0 = (long)blockIdx.x * 256 + threadIdx.x;
  const long stride = (long)gridDim.x * 256;
  float acc = 0.0f;
  for (long i = i0; i < n; i += stride) acc += x[i] * y[i];
  acc = blk_sum<256>(acc, red);
  if (threadIdx.x == 0) pb[blockIdx.x] = acc;
}

__global__ __launch_bounds__(256)
void hgnn_reduce_final(const float* __restrict__ pb, int n, float* __restrict__ out) {
  __shared__ float red[256];
  float acc = 0.0f;
  for (int i = threadIdx.x; i < n; i += 256) acc += pb[i];
  acc = blk_sum<256>(acc, red);
  if (threadIdx.x == 0) out[0] = acc;
}

// ----------------------- spectral norm (power iteration) -------------------
__global__ __launch_bounds__(256)
void hgnn_specnorm(const float* __restrict__ W1, const float* __restrict__ W2,
                   const float* __restrict__ W3, float* __restrict__ out) {
  const float* W = (blockIdx.x == 0) ? W1 : (blockIdx.x == 1 ? W2 : W3);
  const int rows = (blockIdx.x == 2) ? NCLS : HIDW;
  __shared__ float v[HIDW];
  __shared__ float u[HIDW];
  __shared__ float red[256];
  const int tid = threadIdx.x;
  for (int c = tid; c < HIDW; c += 256) v[c] = 1.0f + 0.0009765625f * (float)c;
  __syncthreads();
  float sigma = 0.0f;
  for (int it = 0; it < 40; ++it) {
    float ss = 0.0f;
    for (int c = tid; c < HIDW; c += 256) ss += v[c] * v[c];
    ss = blk_sum<256>(ss, red);
    const float inv = rsqrtf(ss + 1e-30f);
    for (int c = tid; c < HIDW; c += 256) v[c] *= inv;
    __syncthreads();
    for (int r = tid; r < rows; r += 256) {
      const float* wr = W + (size_t)r * HIDW;
      float acc = 0.0f;
      for (int c = 0; c < HIDW; ++c) acc += wr[c] * v[c];
      u[r] = acc;
    }
    __syncthreads();
    float uu = 0.0f;
    for (int r = tid; r < rows; r += 256) uu += u[r] * u[r];
    uu = blk_sum<256>(uu, red);
    sigma = sqrtf(uu);
    for (int c = tid; c < HIDW; c += 256) {
      float acc = 0.0f;
      for (int r = 0; r < rows; ++r) acc += W[(size_t)r * HIDW + c] * u[r];
      v[c] = acc;
    }
    __syncthreads();
  }
  if (tid == 0) out[blockIdx.x] = sigma;
}

// ------------- spectral gap (approximate: shifted power iteration) ---------
// Full eigh(L) is not tractable in-kernel; emit lambda_max - mu_max(shifted)
// as an estimate of the extreme low eigenvalue (single diagnostic scalar).
__global__ __launch_bounds__(1024)
void hgnn_specgap(const float* __restrict__ L, float* __restrict__ out) {
  __shared__ float v[N_NODES];
  __shared__ float u[N_NODES];
  __shared__ float red[1024];
  const int tid = threadIdx.x;
  for (int i = tid; i < N_NODES; i += 1024) v[i] = 1.0f + 0.001f * (float)(i & 7);
  __syncthreads();
  float lmax = 0.0f;
  for (int it = 0; it < 16; ++it) {
    float ss = 0.0f;
    for (int i = tid; i < N_NODES; i += 1024) ss += v[i] * v[i];
    ss = blk_sum<1024>(ss, red);
    const float inv = rsqrtf(ss + 1e-30f);
    for (int i = tid; i < N_NODES; i += 1024) v[i] *= inv;
    __syncthreads();
    for (int r = tid; r < N_NODES; r += 1024) {
      const float* row = L + (size_t)r * N_NODES;
      float acc = 0.0f;
      for (int c = 0; c < N_NODES; ++c) acc += row[c] * v[c];
      u[r] = acc;
    }
    __syncthreads();
    float dd = 0.0f;
    for (int i = tid; i < N_NODES; i += 1024) dd += v[i] * u[i];
    lmax = blk_sum<1024>(dd, red);
    for (int i = tid; i < N_NODES; i += 1024) v[i] = u[i];
    __syncthreads();
  }
  // shifted iteration on (lmax*I - L)
  for (int i = tid; i < N_NODES; i += 1024) v[i] = 1.0f - 0.002f * (float)((i * 37) & 15);
  __syncthreads();
  float mu = 0.0f;
  for (int it = 0; it < 16; ++it) {
    float ss = 0.0f;
    for (int i = tid; i < N_NODES; i += 1024) ss += v[i] * v[i];
    ss = blk_sum<1024>(ss, red);
    const float inv = rsqrtf(ss + 1e-30f);
    for (int i = tid; i < N_NODES; i += 1024) v[i] *= inv;
    __syncthreads();
    for (int r = tid; r < N_NODES; r += 1024) {
      const float* row = L + (size_t)r * N_NODES;
      float acc = 0.0f;
      for (int c = 0; c < N_NODES; ++c) acc += row[c] * v[c];
      u[r] = lmax * v[r] - acc;
    }
    __syncthreads();
    float dd = 0.0f;
    for (int i = tid; i < N_NODES; i += 1024) dd += v[i] * u[i];
    mu = blk_sum<1024>(dd, red);
    for (int i = tid; i < N_NODES; i += 1024) v[i] = u[i];
    __syncthreads();
  }
  if (tid == 0) out[0] = fmaxf(lmax - mu, 0.0f);
}

// ---------------- log_softmax over the NODE axis (axis=1 of [1,N,C]) -------
__global__ __launch_bounds__(256)
void hgnn_logsoftmax(const float* __restrict__ x3, float* __restrict__ out) {
  __shared__ float red[256];
  const int c = blockIdx.x, tid = threadIdx.x;
  float m = -3.0e38f;
  for (int n = tid; n < N_NODES; n += 256) m = fmaxf(m, x3[(size_t)n * NCLS + c]);
  red[tid] = m; __syncthreads();
  #pragma unroll
  for (int s = 128; s > 0; s >>= 1) {
    if (tid < s) red[tid] = fmaxf(red[tid], red[tid + s]);
    __syncthreads();
  }
  m = red[0]; __syncthreads();
  float sum = 0.0f;
  for (int n = tid; n < N_NODES; n += 256) sum += expf(x3[(size_t)n * NCLS + c] - m);
  sum = blk_sum<256>(sum, red);
  const float lse = m + logf(sum);
  for (int n = tid; n < N_NODES; n += 256)
    out[(size_t)n * NCLS + c] = x3[(size_t)n * NCLS + c] - lse;
}

// ---------------------------------------------------------------------------
extern "C" void kernel_launch(void* const* d_in, const int* in_sizes, int n_in,
                              void* d_out, int out_size, void* d_ws, size_t ws_size,
                              hipStream_t stream) {
  (void)in_sizes; (void)n_in; (void)out_size;
  const float* x  = (const float*)d_in[0];
  const float* L  = (const float*)d_in[1];
  const float* H  = (const float*)d_in[2];
  const float* W1 = (const float*)d_in[3];
  const float* W2 = (const float*)d_in[4];
  const float* W3 = (const float*)d_in[5];
  float* out = (float*)d_out;

  float* w = (float*)d_ws;
  size_t off = 0;
  float* z  = w + off; off += (size_t)N_NODES * HIDW;   // L@x scratch
  float* x1 = w + off; off += (size_t)N_NODES * HIDW;
  float* x2 = w + off; off += (size_t)N_NODES * HIDW;
  float* x3 = w + off; off += (size_t)N_NODES * NCLS;
  float* z4 = w + off; off += (size_t)N_NODES * NCLS;
  float* Mb = w + off; off += (size_t)N_EDGES * HIDW;   // H^T y
  float* dv = w + off; off += N_NODES;
  float* rs = w + off; off += N_NODES;                  // 1/sqrt(dv+eps)
  float* de = w + off; off += N_EDGES;
  float* sv = w + off; off += N_NODES;                  // per-node sum(y^2)
  float* av = w + off; off += N_EDGES;                  // H^T s
  float* pb = w + off; off += N_NODES;                  // reduction partials
  if (ws_size < off * sizeof(float)) return;

  float* eL = out + (size_t)N_NODES * NCLS;  // energies_L[3]
  float* eX = eL + 3;                        // energies_expanded[3]
  float* sn = eX + 3;                        // spectral_norms[3]
  float* sg = sn + 3;                        // spectral_gap

  const dim3 blk(256);
  const long nBig = (long)N_NODES * HIDW;
  const long nSml = (long)N_NODES * NCLS;

  hgnn_rowdeg<<<N_NODES, 256, 0, stream>>>(H, dv, rs);
  hgnn_coldeg<<<N_EDGES / 256, 256, 0, stream>>>(H, de);

  // ---- layer 1 ----
  hgnn_wmma_gemm<0,false,4096,512,512,512><<<dim3(32,8), blk, 0, stream>>>(L, x,  z,  nullptr, N_NODES);
  hgnn_wmma_gemm<2,true ,512, 512,512,512><<<dim3(32,8), blk, 0, stream>>>(z, W1, x1, nullptr, IN_CH);
  // z2 = L @ x1 (also feeds energy_L[0])
  hgnn_wmma_gemm<0,false,4096,512,512,512><<<dim3(32,8), blk, 0, stream>>>(L, x1, z,  nullptr, N_NODES);
  hgnn_dot_partial<<<1024, 256, 0, stream>>>(x1, z, nBig, pb);
  hgnn_reduce_final<<<1, 256, 0, stream>>>(pb, 1024, eL + 0);
  hgnn_rowsq<<<N_NODES, 128, 0, stream>>>(x1, rs, sv, HIDW);
  hgnn_wmma_gemm<1,false,2048,512,512,512><<<dim3(16,8), blk, 0, stream>>>(H, x1, Mb, rs, N_NODES);
  hgnn_edge_a<<<N_EDGES / 256, 256, 0, stream>>>(H, sv, av);
  hgnn_edge_energy<<<N_EDGES, 256, 0, stream>>>(Mb, av, de, HIDW, HIDW, pb);
  hgnn_reduce_final<<<1, 256, 0, stream>>>(pb, N_EDGES, eX + 0);

  // ---- layer 2 ----
  hgnn_wmma_gemm<2,true ,512, 512,512,512><<<dim3(32,8), blk, 0, stream>>>(z, W2, x2, nullptr, HIDW);
  hgnn_wmma_gemm<0,false,4096,512,512,512><<<dim3(32,8), blk, 0, stream>>>(L, x2, z,  nullptr, N_NODES);
  hgnn_dot_partial<<<1024, 256, 0, stream>>>(x2, z, nBig, pb);
  hgnn_reduce_final<<<1, 256, 0, stream>>>(pb, 1024, eL + 1);
  hgnn_rowsq<<<N_NODES, 128, 0, stream>>>(x2, rs, sv, HIDW);
  hgnn_wmma_gemm<1,false,2048,512,512,512><<<dim3(16,8), blk, 0, stream>>>(H, x2, Mb, rs, N_NODES);
  hgnn_edge_a<<<N_EDGES / 256, 256, 0, stream>>>(H, sv, av);
  hgnn_edge_energy<<<N_EDGES, 256, 0, stream>>>(Mb, av, de, HIDW, HIDW, pb);
  hgnn_reduce_final<<<1, 256, 0, stream>>>(pb, N_EDGES, eX + 1);

  // ---- layer 3 (NCLS=40, clamped tiles) ----
  hgnn_wmma_gemm<2,true ,512, 512,40, 40 ><<<dim3(32,1), blk, 0, stream>>>(z, W3, x3, nullptr, HIDW);
  hgnn_wmma_gemm<0,false,4096,40, 40, 40 ><<<dim3(32,1), blk, 0, stream>>>(L, x3, z4, nullptr, N_NODES);
  hgnn_dot_partial<<<1024, 256, 0, stream>>>(x3, z4, nSml, pb);
  hgnn_reduce_final<<<1, 256, 0, stream>>>(pb, 1024, eL + 2);
  hgnn_rowsq<<<N_NODES, 128, 0, stream>>>(x3, rs, sv, NCLS);
  hgnn_wmma_gemm<1,false,2048,40, 40, 40 ><<<dim3(16,1), blk, 0, stream>>>(H, x3, Mb, rs, N_NODES);
  hgnn_edge_a<<<N_EDGES / 256, 256, 0, stream>>>(H, sv, av);
  hgnn_edge_energy<<<N_EDGES, 256, 0, stream>>>(Mb, av, de, NCLS, NCLS, pb);
  hgnn_reduce_final<<<1, 256, 0, stream>>>(pb, N_EDGES, eX + 2);

  // ---- diagnostics + output ----
  hgnn_specnorm<<<3, 256, 0, stream>>>(W1, W2, W3, sn);
  hgnn_specgap<<<1, 1024, 0, stream>>>(L, sg);
  hgnn_logsoftmax<<<NCLS, 256, 0, stream>>>(x3, out);
}